// PatchPruning_10857677324657
// MI455X (gfx1250) — compile-verified
//
#include <hip/hip_runtime.h>
#include <stdint.h>

#define B      32
#define N      1024
#define D      768
#define NP     1            // prefix tokens
#define KEEP   512          // N * 0.5
#define OUTROWS (NP + KEEP) // 513
#define ROWS_PER_DESC 16    // TDM gather: up to 16 rows per descriptor (16-bit idx)
#define TILES_TDM ((OUTROWS + ROWS_PER_DESC - 1) / ROWS_PER_DESC) // 33

typedef __attribute__((ext_vector_type(4))) unsigned int u32x4;
typedef __attribute__((ext_vector_type(8))) int          i32x8;
typedef __attribute__((ext_vector_type(4))) int          i32x4;

// ---------------------------------------------------------------------------
// Kernel 1: exact top-K selection per batch, stable-argsort semantics.
// rank(i) = #(v_j > v_i) + #(v_j == v_i && j < i); keep iff rank < KEEP.
// Position of a kept index in ascending order = count of kept indices below
// it (wave32 ballot + per-wave prefix). Writes the ascending keep table to
// workspace and the raw int32 indices into the tail of d_out.
// ---------------------------------------------------------------------------
__global__ void __launch_bounds__(1024)
topk_kernel(const float* __restrict__ mask,
            int* __restrict__ table,      // [B][KEEP] kept src-patch indices, ascending
            int* __restrict__ idx_out)    // raw int32 bits into d_out tail
{
    const int b = blockIdx.x;
    const int i = threadIdx.x;            // 0..1023
    __shared__ float vals[N];
    __shared__ unsigned wcnt[N / 32];

    vals[i] = mask[b * N + i];
    __syncthreads();

    const float v = vals[i];
    int rank = 0;
#pragma unroll 8
    for (int j = 0; j < N; ++j) {
        const float u = vals[j];
        rank += (u > v) || ((u == v) && (j < i));
    }
    const bool keep = (rank < KEEP);

    const int wid  = i >> 5;
    const int lane = i & 31;
    const unsigned bal = __builtin_amdgcn_ballot_w32(keep);
    if (lane == 0) wcnt[wid] = (unsigned)__popc(bal);
    __syncthreads();

    unsigned pre = 0;
    for (int w = 0; w < wid; ++w) pre += wcnt[w];
    const unsigned pos = pre + (unsigned)__popc(bal & ((1u << lane) - 1u));

    if (keep) {
        table[b * KEEP + (int)pos]   = i;
        idx_out[b * KEEP + (int)pos] = i;
    }
}

// ---------------------------------------------------------------------------
// Kernel 2: gather via the CDNA5 Tensor Data Mover.
// One wave per 16-row tile:
//   TDM gather-mode load : 16 indexed rows of x[b] (3 KB each) -> 48 KB LDS
//   TDM normal 2D store  : LDS tile -> 16 contiguous output rows
// Row indices are ascending (prefix row 0, then 1+table[] ascending), as
// required for gather-mode OOB handling. Tracked with TENSORcnt.
// ---------------------------------------------------------------------------
__global__ void __launch_bounds__(32)
gather_tdm_kernel(const float* __restrict__ x,
                  const int* __restrict__ table,
                  float* __restrict__ out)
{
    const int tile = blockIdx.x % TILES_TDM;
    const int b    = blockIdx.x / TILES_TDM;
    const int t    = threadIdx.x;
    const int row_start = tile * ROWS_PER_DESC;
    int nrows = OUTROWS - row_start;
    if (nrows > ROWS_PER_DESC) nrows = ROWS_PER_DESC;

    __shared__ __align__(16) float buf[ROWS_PER_DESC * D]; // 48 KB staging
    __shared__ int srow[ROWS_PER_DESC];

    // Resolve up to 16 source rows (within x[b], incl. prefix row 0).
    if (t < ROWS_PER_DESC) {
        const int row = row_start + t;
        int sr = 0;
        if (row > 0 && row < OUTROWS) sr = 1 + table[b * KEEP + row - 1];
        srow[t] = sr;
    }
    __syncthreads();

    // Pack 16 x 16-bit gather indices into 8 uniform dwords (SGPR-bound).
    uint32_t gi[8];
#pragma unroll
    for (int k = 0; k < 8; ++k) {
        const uint32_t lo = (uint32_t)__builtin_amdgcn_readfirstlane(srow[2 * k]);
        const uint32_t hi = (uint32_t)__builtin_amdgcn_readfirstlane(srow[2 * k + 1]);
        gi[k] = (lo & 0xFFFFu) | (hi << 16);
    }

    const uint32_t lds_off = (uint32_t)(uintptr_t)(&buf[0]);
    const i32x8 gz = {0, 0, 0, 0, 0, 0, 0, 0};    // unused 5th descriptor group

    // ---- TDM gather-mode load: indexed rows -> LDS ----
    {
        const uint64_t ga = (uint64_t)(uintptr_t)(x + (size_t)b * (NP + N) * D);
        u32x4 g0; i32x8 g1; i32x4 g2, g3;
        g0[0] = 0x80000001u;                       // count=1 | gather_mode=1 | 16-bit idx
        g0[1] = lds_off;                           // LDS byte address
        g0[2] = (uint32_t)ga;                      // global_addr[31:0]
        g0[3] = (uint32_t)((ga >> 32) & 0x01FFFFFFu) | 0x80000000u; // addr[56:32] | type=2
        g1[0] = (int)(2u << 16);                   // data_size = 4 bytes
        g1[1] = (int)((uint32_t)D << 16);          // tensor_dim0 = 768 (bits[63:48])
        g1[2] = (int)(((uint32_t)(NP + N) & 0xFFFFu) << 16); // tensor_dim1 = 1025
        g1[3] = (int)((uint32_t)D << 16);          // tile_dim0 = 768 (bits[127:112])
        g1[4] = nrows;                             // tile_dim1 = #valid gather indices
        g1[5] = D;                                 // tensor_dim0_stride = 768
        g1[6] = 0;
        g1[7] = 0;
        g2[0] = (int)gi[0]; g2[1] = (int)gi[1]; g2[2] = (int)gi[2]; g2[3] = (int)gi[3];
        g3[0] = (int)gi[4]; g3[1] = (int)gi[5]; g3[2] = (int)gi[6]; g3[3] = (int)gi[7];
        __builtin_amdgcn_tensor_load_to_lds(g0, g1, g2, g3, gz, 0);
    }

    __builtin_amdgcn_s_wait_tensorcnt(0);          // loads landed in LDS

    // ---- TDM normal-mode 2D store: LDS tile -> contiguous out rows ----
    {
        const uint64_t ga =
            (uint64_t)(uintptr_t)(out + ((size_t)b * OUTROWS + (size_t)row_start) * D);
        u32x4 g0; i32x8 g1; i32x4 g2 = {0, 0, 0, 0}, g3 = {0, 0, 0, 0};
        g0[0] = 0x00000001u;                       // count=1, normal mode
        g0[1] = lds_off;
        g0[2] = (uint32_t)ga;
        g0[3] = (uint32_t)((ga >> 32) & 0x01FFFFFFu) | 0x80000000u; // type=2
        g1[0] = (int)(2u << 16);                   // data_size = 4 bytes
        g1[1] = (int)((uint32_t)D << 16);          // tensor_dim0 = 768
        g1[2] = (int)(((uint32_t)nrows & 0xFFFFu) << 16); // tensor_dim1 = nrows
        g1[3] = (int)((uint32_t)D << 16);          // tile_dim0 = 768
        g1[4] = nrows;                             // tile_dim1 = nrows
        g1[5] = D;                                 // tensor_dim0_stride = 768
        g1[6] = 0;
        g1[7] = 0;
        __builtin_amdgcn_tensor_store_from_lds(g0, g1, g2, g3, gz, 0);
    }

    __builtin_amdgcn_s_wait_tensorcnt(0);          // drain before LDS retires
}

extern "C" void kernel_launch(void* const* d_in, const int* in_sizes, int n_in,
                              void* d_out, int out_size, void* d_ws, size_t ws_size,
                              hipStream_t stream) {
    const float* x    = (const float*)d_in[0];   // [B][NP+N][D] f32
    const float* mask = (const float*)d_in[1];   // [B][N]      f32
    float* out = (float*)d_out;                  // [B][OUTROWS][D] f32, then [B][KEEP] i32
    int*   table   = (int*)d_ws;                 // [B][KEEP] scratch (64 KB)
    int*   idx_out = (int*)(out + (size_t)B * OUTROWS * D);

    topk_kernel<<<B, 1024, 0, stream>>>(mask, table, idx_out);
    gather_tdm_kernel<<<B * TILES_TDM, 32, 0, stream>>>(x, table, out);
}